// GCN_Net_17927193494272
// MI455X (gfx1250) — compile-verified
//
#include <hip/hip_runtime.h>

#define N_NODES 100000
#define N_EDGES 1600000
#define N_PAIRS 500000
#define D       128

typedef __attribute__((ext_vector_type(2))) float v2f;
typedef __attribute__((ext_vector_type(8))) float v8f;

// ---------- degree / normalization ----------
__global__ void k_init_deg(float* __restrict__ deg) {
  int i = blockIdx.x * 256 + threadIdx.x;
  if (i < N_NODES) deg[i] = 1.0f;  // self-loop contributes 1
}

__global__ void k_degree(const int* __restrict__ ei, float* __restrict__ deg) {
  int e = blockIdx.x * 256 + threadIdx.x;
  if (e < N_EDGES) atomicAdd(&deg[ei[N_EDGES + e]], 1.0f);  // dst row
}

__global__ void k_rsqrt(float* __restrict__ deg) {
  int i = blockIdx.x * 256 + threadIdx.x;
  if (i < N_NODES) deg[i] = rsqrtf(deg[i]);  // deg >= 1 always
}

__global__ void k_zero4(float4* __restrict__ p, int n4) {
  int i = blockIdx.x * 256 + threadIdx.x;
  if (i < n4) p[i] = make_float4(0.f, 0.f, 0.f, 0.f);
}

// ---------- dense GEMM via V_WMMA_F32_16X16X4_F32 ----------
// Y[nrows x 128] = X[nrows x 128] @ W[128 x 128]; one wave per 16x16 tile.
__global__ void __launch_bounds__(256) k_gemm(const float* __restrict__ X,
                                              const float* __restrict__ W,
                                              float* __restrict__ Y,
                                              int nrows) {
  const int lane = threadIdx.x & 31;
  const int wv   = threadIdx.x >> 5;
  const int tile = blockIdx.x * 8 + wv;
  const int tm = tile >> 3;       // 8 tiles along N-dim (128/16)
  const int tn = tile & 7;
  if (tm * 16 >= nrows) return;   // wave-uniform; EXEC stays all-1s for WMMA

  const int r    = lane & 15;          // M (for A) / N (for B) within tile
  const int koff = (lane >> 4) << 1;   // lanes 16-31 hold K+2,K+3

  const float* xrow = X + (size_t)(tm * 16 + r) * D;  // A row for this lane
  const float* wcol = W + tn * 16 + r;                // B column for this lane

  v8f acc = {};
#pragma unroll 4
  for (int k = 0; k < D; k += 4) {
    float2 av = *(const float2*)(xrow + k + koff);  // K=k+koff, k+koff+1
    v2f a; a.x = av.x; a.y = av.y;
    v2f b;
    b.x = wcol[(k + koff)     * D];                 // row K=k+koff of W
    b.y = wcol[(k + koff + 1) * D];
    acc = __builtin_amdgcn_wmma_f32_16x16x4_f32(
        /*neg_a=*/false, a, /*neg_b=*/false, b,
        /*c_mod=*/(short)0, acc, /*reuse_a=*/false, /*reuse_b=*/false);
  }

  // C/D layout: VGPR i -> M=i (lanes 0-15) / M=i+8 (lanes 16-31), N = lane&15
  const int rowbase = tm * 16 + ((lane >> 4) << 3);
  const int col     = tn * 16 + r;
#pragma unroll
  for (int i = 0; i < 8; ++i)
    Y[(size_t)(rowbase + i) * D + col] = acc[i];
}

// ---------- edge scatter: agg[dst] += t[src] * dinv[src]*dinv[dst] ----------
// One wave (32 lanes) per edge; float4 per lane covers all 128 features.
__global__ void __launch_bounds__(256) k_aggregate(const int* __restrict__ ei,
                                                   const float* __restrict__ t,
                                                   const float* __restrict__ dinv,
                                                   float* __restrict__ agg) {
  int e = blockIdx.x * 8 + (threadIdx.x >> 5);
  if (e >= N_EDGES) return;
  const int lane = threadIdx.x & 31;
  const int s = ei[e];
  const int d = ei[N_EDGES + e];
  const float nn = dinv[s] * dinv[d];
  float4 v = ((const float4*)(t + (size_t)s * D))[lane];
  float* o = agg + (size_t)d * D + lane * 4;
  atomicAdd(o + 0, v.x * nn);
  atomicAdd(o + 1, v.y * nn);
  atomicAdd(o + 2, v.z * nn);
  atomicAdd(o + 3, v.w * nn);
}

// ---------- add self-loop + bias (+ReLU) ----------
__global__ void k_finish(const float* __restrict__ agg,
                         const float* __restrict__ t,
                         const float* __restrict__ dinv,
                         const float* __restrict__ bias,
                         float* __restrict__ h, int relu) {
  int i = blockIdx.x * 256 + threadIdx.x;
  if (i >= N_NODES * D) return;
  const int node = i >> 7;
  const int d    = i & 127;
  const float di = dinv[node];
  float v = agg[i] + t[i] * di * di + bias[d];
  if (relu) v = fmaxf(v, 0.0f);
  h[i] = v;
}

// ---------- link-prediction dot products ----------
__global__ void __launch_bounds__(256) k_dot(const int* __restrict__ ep,
                                             const int* __restrict__ en,
                                             const float* __restrict__ h,
                                             float* __restrict__ out) {
  int p = blockIdx.x * 8 + (threadIdx.x >> 5);
  if (p >= 2 * N_PAIRS) return;
  const int lane = threadIdx.x & 31;
  const int* pr = (p < N_PAIRS) ? ep : en;
  const int pp  = (p < N_PAIRS) ? p : p - N_PAIRS;
  const int a = pr[2 * pp], b = pr[2 * pp + 1];
  float4 va = ((const float4*)(h + (size_t)a * D))[lane];
  float4 vb = ((const float4*)(h + (size_t)b * D))[lane];
  float s = va.x * vb.x + va.y * vb.y + va.z * vb.z + va.w * vb.w;
#pragma unroll
  for (int m = 16; m > 0; m >>= 1) s += __shfl_xor(s, m, 32);
  if (lane == 0) out[p] = s;
}

extern "C" void kernel_launch(void* const* d_in, const int* in_sizes, int n_in,
                              void* d_out, int out_size, void* d_ws, size_t ws_size,
                              hipStream_t stream) {
  (void)in_sizes; (void)n_in; (void)out_size; (void)ws_size;
  const float* x   = (const float*)d_in[0];
  const float* W1  = (const float*)d_in[1];
  const float* b1  = (const float*)d_in[2];
  const float* W2  = (const float*)d_in[3];
  const float* b2  = (const float*)d_in[4];
  const int* ei    = (const int*)d_in[5];   // [2, N_EDGES]
  const int* edges = (const int*)d_in[6];   // [N_PAIRS, 2]
  const int* edneg = (const int*)d_in[7];   // [N_PAIRS, 2]
  float* out = (float*)d_out;

  const size_t FEAT = (size_t)N_NODES * D;
  float* t    = (float*)d_ws;     // linear-transform output
  float* agg  = t + FEAT;         // neighbor aggregation
  float* h    = agg + FEAT;       // layer output
  float* dinv = h + FEAT;         // deg^{-1/2}

  const int NB_N  = (N_NODES + 255) / 256;
  const int NB_E  = (N_EDGES + 255) / 256;
  const int N4    = (int)(FEAT / 4);
  const int NB_Z  = (N4 + 255) / 256;
  const int NB_F  = (int)((FEAT + 255) / 256);
  const int NB_AG = (N_EDGES + 7) / 8;       // 8 waves / block
  const int NB_GM = (N_NODES / 16) * 8 / 8;  // 50000 tiles, 8 waves / block
  const int NB_DP = (2 * N_PAIRS + 7) / 8;

  // normalization coefficients
  k_init_deg<<<NB_N, 256, 0, stream>>>(dinv);
  k_degree  <<<NB_E, 256, 0, stream>>>(ei, dinv);
  k_rsqrt   <<<NB_N, 256, 0, stream>>>(dinv);

  // layer 1: t = x@W1 ; agg = scatter ; h = relu(agg + t*dinv^2 + b1)
  k_gemm     <<<NB_GM, 256, 0, stream>>>(x, W1, t, N_NODES);
  k_zero4    <<<NB_Z, 256, 0, stream>>>((float4*)agg, N4);
  k_aggregate<<<NB_AG, 256, 0, stream>>>(ei, t, dinv, agg);
  k_finish   <<<NB_F, 256, 0, stream>>>(agg, t, dinv, b1, h, 1);

  // layer 2: t = h@W2 ; agg = scatter ; h = agg + t*dinv^2 + b2
  k_gemm     <<<NB_GM, 256, 0, stream>>>(h, W2, t, N_NODES);
  k_zero4    <<<NB_Z, 256, 0, stream>>>((float4*)agg, N4);
  k_aggregate<<<NB_AG, 256, 0, stream>>>(ei, t, dinv, agg);
  k_finish   <<<NB_F, 256, 0, stream>>>(agg, t, dinv, b2, h, 0);

  // link prediction scores
  k_dot<<<NB_DP, 256, 0, stream>>>(edges, edneg, h, out);
}